// AttnDecoderRNN_74783970558463
// MI455X (gfx1250) — compile-verified
//
#include <hip/hip_runtime.h>
#include <math.h>

// ---------------------------------------------------------------------------
// AttnDecoderRNN single step for MI455X (gfx1250, wave32).
// Memory-bound (~620MB weight traffic, ~27us HBM floor): all large GEMVs are
// coalesced float4 streaming with deterministic tree reductions.
// The attention GEMM h(50x2048) @ Ua^T(2048x50) runs on V_WMMA_F32_16X16X4_F32.
// ---------------------------------------------------------------------------

#define Nh    1024
#define Mv    16000
#define Tt    50
#define TWO_N 2048
#define FOUR_N 4096

// workspace float offsets (total ~1,285,764 floats ~5.2MB)
#define WS_HPAD   0          // 64 x 2048 padded encoder outputs
#define WS_UATP   131072     // 2048 x 64 padded Ua^T
#define WS_C      262144     // context vector (2048)
#define WS_GW     264192     // W^T y   (1024)
#define WS_GWZ    265216
#define WS_GWR    266240
#define WS_UZ     267264     // Uz^T s
#define WS_UR     268288
#define WS_CW     269312     // C^T c
#define WS_CZ     270336
#define WS_CR     271360
#define WS_R      272384
#define WS_Z      273408
#define WS_RS     274432     // r * s
#define WS_URS    275456     // U^T (r*s)
#define WS_VOY    276480     // Vo @ y (4096)
#define WS_TT     280576     // t_tilde (4096)
#define WS_T      284672     // t (2048)
#define WS_LOG    286720     // logits (16000)
#define WS_RED    302720     // {max, log(sumexp)}
#define WS_PART   302724     // reusable partial buffers (<= 3*320*1024)

#define SNEW_OFF  16000      // d_out layout: [log_sm | s_new | alpha]
#define ALPHA_OFF 17024

typedef float v8f __attribute__((ext_vector_type(8)));
typedef float v2f __attribute__((ext_vector_type(2)));

__device__ inline float sigm(float x) { return 1.0f / (1.0f + expf(-x)); }

// deterministic 256-thread block sum; result valid on thread 0
__device__ inline float blockReduce256(float v, float* lds) {
  #pragma unroll
  for (int m = 16; m >= 1; m >>= 1) v += __shfl_xor(v, m, 32);
  int w = threadIdx.x >> 5;
  if ((threadIdx.x & 31) == 0) lds[w] = v;
  __syncthreads();
  float r = 0.0f;
  if (threadIdx.x == 0) {
    for (int i = 0; i < 8; ++i) r += lds[i];
  }
  return r;
}

// ---------------------------------------------------------------------------
// Pad h -> hpad(64x2048) and transpose+pad Ua(50x2048) -> uatp(2048x64)
// ---------------------------------------------------------------------------
__global__ void pad_kernel(const float* __restrict__ h, const float* __restrict__ Ua,
                           float* __restrict__ hpad, float* __restrict__ uatp) {
  int idx = blockIdx.x * blockDim.x + threadIdx.x;
  if (idx < 64 * TWO_N) {
    int t = idx / TWO_N, k = idx % TWO_N;
    hpad[idx] = (t < Tt) ? h[t * TWO_N + k] : 0.0f;
  } else {
    int i2 = idx - 64 * TWO_N;      // k*64 + np
    int k = i2 / 64, np = i2 % 64;
    uatp[i2] = (np < Tt) ? Ua[np * TWO_N + k] : 0.0f;
  }
}

// ---------------------------------------------------------------------------
// Attention: WMMA 16x16x4 f32 GEMM + softmax + alpha + context. One WG.
// ---------------------------------------------------------------------------
__global__ __launch_bounds__(512) void attn_kernel(
    const float* __restrict__ s, const float* __restrict__ Wa,
    const float* __restrict__ Va, const float* __restrict__ henc,
    const float* __restrict__ hpad, const float* __restrict__ uatp,
    float* __restrict__ cvec, float* __restrict__ outp) {
  __shared__ float s_sh[Nh];
  __shared__ float was_sh[64];
  __shared__ float va_sh[64];
  __shared__ float part_sh[4 * 64];
  __shared__ float scores_sh[64];
  __shared__ float attn_sh[64];

  const int tid  = threadIdx.x;
  const int wv   = tid >> 5;     // 0..15
  const int lane = tid & 31;
  const int half = lane >> 4;
  const int l16  = lane & 15;

  for (int i = tid; i < Nh; i += 512) s_sh[i] = s[i];
  if (tid < 64) va_sh[tid] = (tid < Tt) ? Va[tid] : 0.0f;
  __syncthreads();

  // was[np] = Wa[np,:] . s   (one wave per 4 np values)
  #pragma unroll
  for (int i = 0; i < 4; ++i) {
    int np = wv * 4 + i;
    float p = 0.0f;
    if (np < Tt) {
      for (int k = lane; k < Nh; k += 32) p += Wa[np * Nh + k] * s_sh[k];
    }
    #pragma unroll
    for (int m = 16; m >= 1; m >>= 1) p += __shfl_xor(p, m, 32);
    if (lane == 0) was_sh[np] = p;
  }
  __syncthreads();

  // E = hpad @ uatp via WMMA f32 16x16x4; wave wv owns tile (ti,nj)
  const int ti = wv >> 2, nj = wv & 3;
  const int arow = ti * 16 + l16;
  const int bcol = nj * 16 + l16;
  v8f acc = {0.f, 0.f, 0.f, 0.f, 0.f, 0.f, 0.f, 0.f};
  for (int k = 0; k < TWO_N; k += 4) {
    const int ka = k + half * 2;
    v2f a, b;
    a.x = hpad[arow * TWO_N + ka];
    a.y = hpad[arow * TWO_N + ka + 1];
    b.x = uatp[ka * 64 + bcol];
    b.y = uatp[(ka + 1) * 64 + bcol];
    acc = __builtin_amdgcn_wmma_f32_16x16x4_f32(false, a, false, b,
                                                (short)0, acc, false, false);
  }

  // scores partial: sum over this tile's cols of tanh(E + was[col]) * Va[col]
  const float wascol = was_sh[bcol];
  const float vacol  = va_sh[bcol];
  #pragma unroll
  for (int v = 0; v < 8; ++v) {
    float e = tanhf(acc[v] + wascol) * vacol;
    #pragma unroll
    for (int m = 1; m < 16; m <<= 1) e += __shfl_xor(e, m, 32);  // within 16-lane half
    if (l16 == 0) part_sh[nj * 64 + ti * 16 + v + half * 8] = e;
  }
  __syncthreads();

  if (tid < 64) {
    scores_sh[tid] = part_sh[tid] + part_sh[64 + tid] +
                     part_sh[128 + tid] + part_sh[192 + tid];
  }
  __syncthreads();

  // softmax over T=50 (serial, deterministic, trivial size)
  if (tid == 0) {
    float mx = -3.4e38f;
    for (int t = 0; t < Tt; ++t) mx = fmaxf(mx, scores_sh[t]);
    float ssum = 0.0f;
    for (int t = 0; t < Tt; ++t) { float e = expf(scores_sh[t] - mx); attn_sh[t] = e; ssum += e; }
    float inv = 1.0f / ssum;
    for (int t = 0; t < Tt; ++t) attn_sh[t] *= inv;
  }
  __syncthreads();

  // alpha = h * attn  (to d_out) and c = sum_t alpha  (to ws)
  const int j4 = tid * 4;  // 512 threads * 4 = 2048 cols
  float4 csum = {0.f, 0.f, 0.f, 0.f};
  for (int t = 0; t < Tt; ++t) {
    float av = attn_sh[t];
    float4 hv = *(const float4*)&henc[t * TWO_N + j4];
    float4 al = make_float4(hv.x * av, hv.y * av, hv.z * av, hv.w * av);
    *(float4*)&outp[ALPHA_OFF + t * TWO_N + j4] = al;
    csum.x += al.x; csum.y += al.y; csum.z += al.z; csum.w += al.w;
  }
  *(float4*)&cvec[j4] = csum;
}

// ---------------------------------------------------------------------------
// A^T x for up to 3 row-major (rows x 1024) matrices: column-parallel partials.
// Block b accumulates rows [b*rpb, ...) into P[b*1024 + col]. Coalesced f4.
// ---------------------------------------------------------------------------
__global__ __launch_bounds__(256) void gemvT_part3(
    const float* __restrict__ A0, const float* __restrict__ A1,
    const float* __restrict__ A2, const float* __restrict__ x,
    float* __restrict__ P0, float* __restrict__ P1, float* __restrict__ P2,
    int rowsPerBlock, int totalRows) {
  const int b = blockIdx.x;
  int r0 = b * rowsPerBlock;
  int r1 = r0 + rowsPerBlock; if (r1 > totalRows) r1 = totalRows;
  const int c4 = threadIdx.x * 4;
  float4 a0 = {0,0,0,0}, a1 = {0,0,0,0}, a2 = {0,0,0,0};
  for (int m = r0; m < r1; ++m) {
    const float xm = x[m];
    const size_t off = (size_t)m * Nh + c4;
    float4 w0 = *(const float4*)(A0 + off);
    a0.x += xm * w0.x; a0.y += xm * w0.y; a0.z += xm * w0.z; a0.w += xm * w0.w;
    if (A1) {
      float4 w1 = *(const float4*)(A1 + off);
      a1.x += xm * w1.x; a1.y += xm * w1.y; a1.z += xm * w1.z; a1.w += xm * w1.w;
    }
    if (A2) {
      float4 w2 = *(const float4*)(A2 + off);
      a2.x += xm * w2.x; a2.y += xm * w2.y; a2.z += xm * w2.z; a2.w += xm * w2.w;
    }
  }
  const size_t po = (size_t)b * Nh + c4;
  *(float4*)(P0 + po) = a0;
  if (A1) *(float4*)(P1 + po) = a1;
  if (A2) *(float4*)(P2 + po) = a2;
}

__global__ void gemvT_reduce3(const float* __restrict__ P0, const float* __restrict__ P1,
                              const float* __restrict__ P2, float* __restrict__ O0,
                              float* __restrict__ O1, float* __restrict__ O2, int nb) {
  const int col = blockIdx.x * blockDim.x + threadIdx.x;
  if (col >= Nh) return;
  float s0 = 0, s1 = 0, s2 = 0;
  for (int b = 0; b < nb; ++b) {
    const size_t o = (size_t)b * Nh + col;
    s0 += P0[o];
    if (P1) s1 += P1[o];
    if (P2) s2 += P2[o];
  }
  O0[col] = s0;
  if (O1) O1[col] = s1;
  if (O2) O2[col] = s2;
}

// ---------------------------------------------------------------------------
// Row-dot GEMV: one block per row, contiguous float4 stream. len4 = cols/4.
// ---------------------------------------------------------------------------
__global__ __launch_bounds__(256) void rowdot1(const float* __restrict__ A,
                                               const float* __restrict__ x,
                                               float* __restrict__ out, int len4) {
  __shared__ float lds[8];
  const float4* row = (const float4*)(A + (size_t)blockIdx.x * ((size_t)len4 * 4));
  const float4* xv4 = (const float4*)x;
  float acc = 0.0f;
  for (int i = threadIdx.x; i < len4; i += 256) {
    float4 a = row[i];
    float4 xv = xv4[i];
    acc += a.x * xv.x + a.y * xv.y + a.z * xv.z + a.w * xv.w;
  }
  float r = blockReduce256(acc, lds);
  if (threadIdx.x == 0) out[blockIdx.x] = r;
}

// t_tilde[row] = sigmoid(Uo[row,:].s_new + Co[row,:].c + voy[row])
__global__ __launch_bounds__(256) void ttilde_kernel(
    const float* __restrict__ Uo, const float* __restrict__ Co,
    const float* __restrict__ snew, const float* __restrict__ cvec,
    const float* __restrict__ voy, float* __restrict__ tt) {
  __shared__ float lds[8];
  const int row = blockIdx.x;
  float acc = 0.0f;
  const float4* uo4 = (const float4*)(Uo + (size_t)row * Nh);
  const float4* sn4 = (const float4*)snew;
  for (int i = threadIdx.x; i < Nh / 4; i += 256) {
    float4 a = uo4[i]; float4 b = sn4[i];
    acc += a.x * b.x + a.y * b.y + a.z * b.z + a.w * b.w;
  }
  const float4* co4 = (const float4*)(Co + (size_t)row * TWO_N);
  const float4* c4 = (const float4*)cvec;
  for (int i = threadIdx.x; i < TWO_N / 4; i += 256) {
    float4 a = co4[i]; float4 b = c4[i];
    acc += a.x * b.x + a.y * b.y + a.z * b.z + a.w * b.w;
  }
  float r = blockReduce256(acc, lds);
  if (threadIdx.x == 0) tt[row] = sigm(r + voy[row]);
}

__global__ void rz_kernel(const float* __restrict__ gWr, const float* __restrict__ ur,
                          const float* __restrict__ cr, const float* __restrict__ gWz,
                          const float* __restrict__ uz, const float* __restrict__ cz,
                          const float* __restrict__ s, float* __restrict__ R,
                          float* __restrict__ Z, float* __restrict__ RS) {
  const int i = blockIdx.x * blockDim.x + threadIdx.x;
  if (i >= Nh) return;
  float rv = sigm(gWr[i] + ur[i] + cr[i]);
  float zv = sigm(gWz[i] + uz[i] + cz[i]);
  R[i] = rv; Z[i] = zv; RS[i] = rv * s[i];
}

__global__ void snew_kernel(const float* __restrict__ gW, const float* __restrict__ urs,
                            const float* __restrict__ cw, const float* __restrict__ Z,
                            const float* __restrict__ s, float* __restrict__ outp) {
  const int i = blockIdx.x * blockDim.x + threadIdx.x;
  if (i >= Nh) return;
  float st = sigm(gW[i] + urs[i] + cw[i]);
  outp[SNEW_OFF + i] = (1.0f - Z[i]) * s[i] + Z[i] * st;
}

__global__ void tmax_kernel(const float* __restrict__ tt, float* __restrict__ t) {
  const int j = blockIdx.x * blockDim.x + threadIdx.x;
  if (j >= TWO_N) return;
  t[j] = fmaxf(tt[2 * j], tt[2 * j + 1]);
}

__global__ __launch_bounds__(1024) void lsm_reduce(const float* __restrict__ logits,
                                                   float* __restrict__ red) {
  __shared__ float lds[32];
  __shared__ float bc;
  const int tid = threadIdx.x;
  float mx = -3.4e38f;
  for (int i = tid; i < Mv; i += 1024) mx = fmaxf(mx, logits[i]);
  #pragma unroll
  for (int m = 16; m >= 1; m >>= 1) mx = fmaxf(mx, __shfl_xor(mx, m, 32));
  if ((tid & 31) == 0) lds[tid >> 5] = mx;
  __syncthreads();
  if (tid == 0) {
    float m2 = lds[0];
    for (int i = 1; i < 32; ++i) m2 = fmaxf(m2, lds[i]);
    bc = m2;
  }
  __syncthreads();
  const float gmax = bc;
  float sm = 0.0f;
  for (int i = tid; i < Mv; i += 1024) sm += expf(logits[i] - gmax);
  #pragma unroll
  for (int m = 16; m >= 1; m >>= 1) sm += __shfl_xor(sm, m, 32);
  __syncthreads();
  if ((tid & 31) == 0) lds[tid >> 5] = sm;
  __syncthreads();
  if (tid == 0) {
    float t = 0.0f;
    for (int i = 0; i < 32; ++i) t += lds[i];
    red[0] = gmax; red[1] = logf(t);
  }
}

__global__ void lsm_write(const float* __restrict__ logits, const float* __restrict__ red,
                          float* __restrict__ outp) {
  const int i = blockIdx.x * blockDim.x + threadIdx.x;
  if (i >= Mv) return;
  outp[i] = logits[i] - red[0] - red[1];
}

// ---------------------------------------------------------------------------
extern "C" void kernel_launch(void* const* d_in, const int* in_sizes, int n_in,
                              void* d_out, int out_size, void* d_ws, size_t ws_size,
                              hipStream_t stream) {
  const float* y  = (const float*)d_in[0];
  const float* s  = (const float*)d_in[1];
  const float* h  = (const float*)d_in[2];
  const float* W  = (const float*)d_in[3];
  const float* Wz = (const float*)d_in[4];
  const float* Wr = (const float*)d_in[5];
  const float* Wo = (const float*)d_in[6];
  const float* U  = (const float*)d_in[7];
  const float* Uz = (const float*)d_in[8];
  const float* Ur = (const float*)d_in[9];
  const float* Uo = (const float*)d_in[10];
  const float* C  = (const float*)d_in[11];
  const float* Cz = (const float*)d_in[12];
  const float* Cr = (const float*)d_in[13];
  const float* Co = (const float*)d_in[14];
  const float* Vo = (const float*)d_in[15];
  const float* Va = (const float*)d_in[16];
  const float* Wa = (const float*)d_in[17];
  const float* Ua = (const float*)d_in[18];
  float* outp = (float*)d_out;
  float* ws = (float*)d_ws;

  float* P0 = ws + WS_PART;
  float* P1 = P0 + 320 * Nh;
  float* P2 = P1 + 320 * Nh;

  // 1) pad h / Ua^T for WMMA fragments
  pad_kernel<<<1024, 256, 0, stream>>>(h, Ua, ws + WS_HPAD, ws + WS_UATP);

  // 2) y-dependent GEMVs (independent of attention)
  gemvT_part3<<<320, 256, 0, stream>>>(W, Wz, Wr, y, P0, P1, P2, 50, Mv);
  gemvT_reduce3<<<4, 256, 0, stream>>>(P0, P1, P2, ws + WS_GW, ws + WS_GWZ,
                                       ws + WS_GWR, 320);
  gemvT_part3<<<32, 256, 0, stream>>>(Uz, Ur, nullptr, s, P0, P1, nullptr, 32, Nh);
  gemvT_reduce3<<<4, 256, 0, stream>>>(P0, P1, nullptr, ws + WS_UZ, ws + WS_UR,
                                       nullptr, 32);
  rowdot1<<<FOUR_N, 256, 0, stream>>>(Vo, y, ws + WS_VOY, Mv / 4);

  // 3) attention (WMMA) -> attn, alpha (d_out), context c (ws)
  attn_kernel<<<1, 512, 0, stream>>>(s, Wa, Va, h, ws + WS_HPAD, ws + WS_UATP,
                                     ws + WS_C, outp);

  // 4) context-dependent GEMVs, gates, s_new
  gemvT_part3<<<64, 256, 0, stream>>>(C, Cz, Cr, ws + WS_C, P0, P1, P2, 32, TWO_N);
  gemvT_reduce3<<<4, 256, 0, stream>>>(P0, P1, P2, ws + WS_CW, ws + WS_CZ,
                                       ws + WS_CR, 64);
  rz_kernel<<<4, 256, 0, stream>>>(ws + WS_GWR, ws + WS_UR, ws + WS_CR,
                                   ws + WS_GWZ, ws + WS_UZ, ws + WS_CZ,
                                   s, ws + WS_R, ws + WS_Z, ws + WS_RS);
  gemvT_part3<<<32, 256, 0, stream>>>(U, nullptr, nullptr, ws + WS_RS, P0, nullptr,
                                      nullptr, 32, Nh);
  gemvT_reduce3<<<4, 256, 0, stream>>>(P0, nullptr, nullptr, ws + WS_URS, nullptr,
                                       nullptr, 32);
  snew_kernel<<<4, 256, 0, stream>>>(ws + WS_GW, ws + WS_URS, ws + WS_CW,
                                     ws + WS_Z, s, outp);

  // 5) maxout output layer + log-softmax
  ttilde_kernel<<<FOUR_N, 256, 0, stream>>>(Uo, Co, outp + SNEW_OFF, ws + WS_C,
                                            ws + WS_VOY, ws + WS_TT);
  tmax_kernel<<<8, 256, 0, stream>>>(ws + WS_TT, ws + WS_T);
  rowdot1<<<Mv, 256, 0, stream>>>(Wo, ws + WS_T, ws + WS_LOG, TWO_N / 4);
  lsm_reduce<<<1, 1024, 0, stream>>>(ws + WS_LOG, ws + WS_RED);
  lsm_write<<<(Mv + 255) / 256, 256, 0, stream>>>(ws + WS_LOG, ws + WS_RED, outp);
}